// GraphConvolution_74861279969816
// MI455X (gfx1250) — compile-verified
//
#include <hip/hip_runtime.h>
#include <hip/hip_bf16.h>

typedef __attribute__((ext_vector_type(2))) float v2f;
typedef __attribute__((ext_vector_type(4))) float v4f;
typedef __attribute__((ext_vector_type(8))) float v8f;

#define N_NODES 50000
#define FDIM    256

// ---------------------------------------------------------------------------
// Kernel 1: zero the aggregation workspace (d_ws is poisoned by the harness).
// One float4 per thread, 12.8M floats -> 3.2M float4 -> 12500 blocks x 256.
// ---------------------------------------------------------------------------
__global__ void gc_zero_kernel(float* __restrict__ agg) {
    size_t i = ((size_t)blockIdx.x * blockDim.x + threadIdx.x) * 4;
    v4f z = {};
    *(v4f*)(agg + i) = z;
}

// ---------------------------------------------------------------------------
// Kernel 2: edge-parallel SpMM scatter with fp32 global atomics.
// One wave (32 lanes) per edge; lane handles features [lane*8, lane*8+8).
// x row gather = 1 KB coalesced per wave; 8 global_atomic_add_f32 per lane.
// x (51.2 MB) and agg (51.2 MB) both live in the 192 MB L2, so the scatter
// is L2-atomic-bound rather than HBM-bound.
// ---------------------------------------------------------------------------
__global__ void gc_spmm_kernel(const float* __restrict__ x,
                               const int*   __restrict__ erow,
                               const int*   __restrict__ ecol,
                               const float* __restrict__ eval,
                               float*       __restrict__ agg,
                               int E) {
    int wave = blockIdx.x * (blockDim.x >> 5) + (threadIdx.x >> 5);
    int lane = threadIdx.x & 31;
    if (wave >= E) return;

    int   r = erow[wave];
    int   c = ecol[wave];
    float v = eval[wave];

    const float* xr = x   + (size_t)c * FDIM + lane * 8;
    float*       ar = agg + (size_t)r * FDIM + lane * 8;

    v4f x0 = *(const v4f*)(xr);
    v4f x1 = *(const v4f*)(xr + 4);

#pragma unroll
    for (int j = 0; j < 4; ++j) {
        (void)__hip_atomic_fetch_add(ar + j, v * x0[j],
                                     __ATOMIC_RELAXED, __HIP_MEMORY_SCOPE_AGENT);
    }
#pragma unroll
    for (int j = 0; j < 4; ++j) {
        (void)__hip_atomic_fetch_add(ar + 4 + j, v * x1[j],
                                     __ATOMIC_RELAXED, __HIP_MEMORY_SCOPE_AGENT);
    }
}

// ---------------------------------------------------------------------------
// Kernel 3: out = agg @ W^T + b with V_WMMA_F32_16X16X4_F32 (full fp32).
//
// Wave tile: 16 node-rows x 64 out-features = 4 accumulator tiles, so the
// A fragment is loaded once and fed to 4 WMMAs per k-step:
//   per k-step: 1 A b64 load + 4 B b64 loads + 4 WMMAs (1.25 loads/WMMA).
//
// fp32 A layout (16x4): lanes 0-15 hold K={k,k+1}, lanes 16-31 hold K={k+2,k+3}
//   -> per-lane A fragment = contiguous float2 at agg[row*256 + k + 2*half].
// B = W^T, so B[k][n] = W[n][k]; B fragment = contiguous float2 from
//   W[(o_base+n)*256 + k + 2*half]. No transpose staging needed.
// C/D layout: VGPR j -> rows {j, j+8} (selected by lane half), col = lane%16.
// ---------------------------------------------------------------------------
__global__ void gc_gemm_wmma_kernel(const float* __restrict__ agg,
                                    const float* __restrict__ W,
                                    const float* __restrict__ bias,
                                    float*       __restrict__ out) {
    int lane = threadIdx.x & 31;
    int wave = threadIdx.x >> 5;          // 0..3

    int row_base = blockIdx.x << 4;       // 16 node rows per block
    int o_base   = wave << 6;             // 64 out features per wave
    int half     = lane >> 4;             // 0: lanes 0-15, 1: lanes 16-31
    int l16      = lane & 15;

    const float* aptr = agg + (size_t)(row_base + l16) * FDIM + (half << 1);
    const float* bptr = W   + (size_t)(o_base  + l16) * FDIM + (half << 1);

    v8f acc[4] = {{}, {}, {}, {}};

#pragma unroll 4
    for (int k = 0; k < FDIM; k += 4) {
        v2f av = *(const v2f*)(aptr + k);
#pragma unroll
        for (int t = 0; t < 4; ++t) {
            v2f bv = *(const v2f*)(bptr + (size_t)t * 16 * FDIM + k);
            acc[t] = __builtin_amdgcn_wmma_f32_16x16x4_f32(
                /*neg_a=*/false, av, /*neg_b=*/false, bv,
                /*c_mod=*/(short)0, acc[t], /*reuse_a=*/false, /*reuse_b=*/false);
        }
    }

#pragma unroll
    for (int t = 0; t < 4; ++t) {
        int   ocol = o_base + t * 16 + l16;
        float bv   = bias[ocol];
        float* op  = out + (size_t)(row_base + (half << 3)) * FDIM + ocol;
#pragma unroll
        for (int j = 0; j < 8; ++j) {
            op[(size_t)j * FDIM] = acc[t][j] + bv;
        }
    }
}

// ---------------------------------------------------------------------------
// Launch: zero -> spmm -> gemm, all on `stream` (graph-capture safe).
// Inputs: x[50000*256] f32, edge_row[E] i32, edge_col[E] i32, edge_val[E] f32,
//         W[256*256] f32, b[256] f32.  Output: [50000,256] f32.
// Workspace: agg = 50000*256*4 = 51.2 MB in d_ws.
// ---------------------------------------------------------------------------
extern "C" void kernel_launch(void* const* d_in, const int* in_sizes, int n_in,
                              void* d_out, int out_size, void* d_ws, size_t ws_size,
                              hipStream_t stream) {
    const float* x    = (const float*)d_in[0];
    const int*   erow = (const int*)  d_in[1];
    const int*   ecol = (const int*)  d_in[2];
    const float* eval = (const float*)d_in[3];
    const float* W    = (const float*)d_in[4];
    const float* b    = (const float*)d_in[5];
    float*       out  = (float*)d_out;
    float*       agg  = (float*)d_ws;

    const int E = in_sizes[1];

    // 1) zero agg: 50000*256 floats = 3,200,000 float4
    gc_zero_kernel<<<(N_NODES * FDIM / 4 + 255) / 256, 256, 0, stream>>>(agg);

    // 2) SpMM scatter: one wave per edge, 8 waves per 256-thread block
    int blocks = (E + 7) / 8;
    gc_spmm_kernel<<<blocks, 256, 0, stream>>>(x, erow, ecol, eval, agg, E);

    // 3) WMMA GEMM: 3125 row-tile blocks x 4 waves (64 cols each) = 256 cols
    gc_gemm_wmma_kernel<<<N_NODES / 16, 128, 0, stream>>>(agg, W, b, out);
}